// InfluenceSpreadNN_1176821039627
// MI455X (gfx1250) — compile-verified
//
#include <hip/hip_runtime.h>
#include <cstdint>

// ---------------------------------------------------------------------------
// InfluenceSpread: 5 rounds of gather (p[src]) -> mul edge_prob -> scatter-add
// (atomic f32 into agg[tgt]) -> pointwise node update.
//
// Bandwidth-bound: 384 MB of edge data streamed per step (x5). Node arrays
// (p/prod/agg, 2 MB each) stay resident in the 192 MB L2. Edge streams are
// staged via CDNA5 async global->LDS copies (ASYNCcnt) with per-wave double
// buffering; scatter uses native global_atomic_add_f32 (relaxed, agent scope,
// non-returning).
// ---------------------------------------------------------------------------

#define AS1 __attribute__((address_space(1)))
#define AS3 __attribute__((address_space(3)))

typedef __attribute__((ext_vector_type(4))) int   int4v;
typedef __attribute__((ext_vector_type(4))) float float4v;

// Exact pointer types the async-LDS builtin expects (from hipcc diagnostic):
// arg0: pointer to int __attribute__((vector_size(16))) in addrspace(1),
// arg1: same vector type in addrspace(3).
typedef int v4i __attribute__((vector_size(16)));
typedef AS1 v4i* g4p;   // global (AS1) pointer to 16-byte vector
typedef AS3 v4i* l4p;   // LDS   (AS3) pointer to 16-byte vector

#define THREADS   256
#define TILE_E    256                 // edges per wave tile
#define BUF_BYTES (TILE_E * 12)       // prob[1KB] | src[1KB] | tgt[1KB]
#define WAVE_LDS  (2 * BUF_BYTES)     // double buffered

#if defined(__HIP_DEVICE_COMPILE__) && \
    __has_builtin(__builtin_amdgcn_global_load_async_to_lds_b128) && \
    __has_builtin(__builtin_amdgcn_s_wait_asynccnt)
#define USE_ASYNC_LDS 1
#else
#define USE_ASYNC_LDS 0
#endif

#define COMPILER_BARRIER() asm volatile("" ::: "memory")

__device__ __forceinline__ void atomic_add_f32_agent(float* p, float v) {
  // Relaxed agent-scope FP32 add -> global_atomic_add_f32 (no return needed).
  (void)__hip_atomic_fetch_add(p, v, __ATOMIC_RELAXED, __HIP_MEMORY_SCOPE_AGENT);
}

// ---------------------------------------------------------------------------
// Edge kernel: agg[tgt[e]] += edge_prob[e] * p[src[e]]  for all e.
// ---------------------------------------------------------------------------
__global__ __launch_bounds__(THREADS)
void edge_scatter(const int* __restrict__ src, const int* __restrict__ tgt,
                  const float* __restrict__ ep, const float* __restrict__ p,
                  float* __restrict__ agg, long long E) {
  const long long gtid = (long long)blockIdx.x * THREADS + threadIdx.x;
  const long long nthreads = (long long)gridDim.x * THREADS;

#if USE_ASYNC_LDS
  __shared__ char smem[(THREADS / 32) * WAVE_LDS];
  const int  lane  = threadIdx.x & 31;
  const int  lwave = threadIdx.x >> 5;
  char* const my   = &smem[lwave * WAVE_LDS];

  const long long wid    = gtid >> 5;
  const long long nwaves = nthreads >> 5;
  const long long tiles  = E / TILE_E;

  // Stage one 256-edge tile (three streams) into LDS buffer `buf` via
  // async global->LDS B128 copies: 2 ops/stream, 512B per op per wave.
  auto fill = [&](int buf, long long tile) {
    const long long base = tile * (long long)TILE_E * 4;  // byte offset
    char* const lb = my + buf * BUF_BYTES;
#pragma unroll
    for (int k = 0; k < 2; ++k) {
      const int off = lane * 16 + k * 512;
      __builtin_amdgcn_global_load_async_to_lds_b128(
          (g4p)((const char*)ep + base + off),
          (l4p)(lb + off), 0, 0);
      __builtin_amdgcn_global_load_async_to_lds_b128(
          (g4p)((const char*)src + base + off),
          (l4p)(lb + 1024 + off), 0, 0);
      __builtin_amdgcn_global_load_async_to_lds_b128(
          (g4p)((const char*)tgt + base + off),
          (l4p)(lb + 2048 + off), 0, 0);
    }
  };

  // Consume a staged tile: strided lane access (lane + 32*k) is LDS
  // bank-conflict-free; gather p[src] hits L2, scatter is a non-returning
  // f32 atomic (STOREcnt -> never stalls this loop).
  auto process = [&](int buf) {
    char* const lb = my + buf * BUF_BYTES;
    const float* pb = (const float*)lb;
    const int*   sb = (const int*)(lb + 1024);
    const int*   tb = (const int*)(lb + 2048);
#pragma unroll
    for (int k = 0; k < 8; ++k) {
      const int i = lane + 32 * k;
      const float v = pb[i] * p[sb[i]];
      atomic_add_f32_agent(&agg[tb[i]], v);
    }
  };

  long long t = wid;
  int cur = 0;
  if (t < tiles) fill(0, t);
  for (; t < tiles; t += nwaves) {
    const long long tn = t + nwaves;
    if (tn < tiles) {
      fill(cur ^ 1, tn);                       // prefetch next tile (6 ops)
      __builtin_amdgcn_s_wait_asynccnt(6);     // current tile's 6 ops done
    } else {
      __builtin_amdgcn_s_wait_asynccnt(0);
    }
    COMPILER_BARRIER();
    process(cur);
    cur ^= 1;
  }

  // Tail edges (E not a multiple of TILE_E): plain scalar path.
  for (long long e = tiles * TILE_E + gtid; e < E; e += nthreads) {
    atomic_add_f32_agent(&agg[tgt[e]], ep[e] * p[src[e]]);
  }
#else
  // Fallback: 4-wide non-temporal streaming loads (keep node arrays in L2).
  const long long quads = E >> 2;
  for (long long q = gtid; q < quads; q += nthreads) {
    const int4v   s4 = __builtin_nontemporal_load((const int4v*)src + q);
    const int4v   t4 = __builtin_nontemporal_load((const int4v*)tgt + q);
    const float4v e4 = __builtin_nontemporal_load((const float4v*)ep + q);
    atomic_add_f32_agent(&agg[t4.x], e4.x * p[s4.x]);
    atomic_add_f32_agent(&agg[t4.y], e4.y * p[s4.y]);
    atomic_add_f32_agent(&agg[t4.z], e4.z * p[s4.z]);
    atomic_add_f32_agent(&agg[t4.w], e4.w * p[s4.w]);
  }
  for (long long e = (quads << 2) + gtid; e < E; e += nthreads) {
    atomic_add_f32_agent(&agg[tgt[e]], ep[e] * p[src[e]]);
  }
#endif
}

// ---------------------------------------------------------------------------
// Node kernels
// ---------------------------------------------------------------------------
__global__ __launch_bounds__(THREADS)
void init_state(const float* __restrict__ prior, float* __restrict__ p,
                float* __restrict__ prod, float* __restrict__ agg, int n) {
  const int i = blockIdx.x * blockDim.x + threadIdx.x;
  if (i >= n) return;
  p[i]    = prior[i];
  prod[i] = 1.0f;
  agg[i]  = 0.0f;
}

__global__ __launch_bounds__(THREADS)
void node_update(float* __restrict__ p, float* __restrict__ prod,
                 float* __restrict__ agg, const float* __restrict__ prior,
                 float* __restrict__ out, int n, int last) {
  const int i = blockIdx.x * blockDim.x + threadIdx.x;
  if (i >= n) return;
  const float a = agg[i];
  agg[i] = 0.0f;                       // reset for next step's scatter
  const float pr    = prod[i];
  const float delta = 1.0f - __expf(-a);
  const float np    = pr * delta;      // new p_t_u
  const float prn   = pr * (1.0f - np);
  p[i]    = np;
  prod[i] = prn;
  if (last) out[i] = 1.0f - prn + prior[i];
}

// ---------------------------------------------------------------------------
// Launch: init, then 5 x (edge scatter -> node update); last update writes out.
// ---------------------------------------------------------------------------
extern "C" void kernel_launch(void* const* d_in, const int* in_sizes, int n_in,
                              void* d_out, int out_size, void* d_ws, size_t ws_size,
                              hipStream_t stream) {
  const float* prior = (const float*)d_in[0];
  const int*   eidx  = (const int*)d_in[1];
  const float* ep    = (const float*)d_in[2];
  const int       N  = in_sizes[0];
  const long long E  = (long long)in_sizes[2];
  const int* src = eidx;       // edge_index row 0
  const int* tgt = eidx + E;   // edge_index row 1

  float* p    = (float*)d_ws;          // N
  float* prod = p + N;                 // N
  float* agg  = prod + N;              // N  (3*N*4 = 6 MB total)
  float* out  = (float*)d_out;

  const int nodeBlocks = (N + THREADS - 1) / THREADS;
  const int edgeBlocks = 4096;         // grid-stride over edge tiles

  init_state<<<nodeBlocks, THREADS, 0, stream>>>(prior, p, prod, agg, N);
  for (int s = 0; s < 5; ++s) {
    edge_scatter<<<edgeBlocks, THREADS, 0, stream>>>(src, tgt, ep, p, agg, E);
    node_update<<<nodeBlocks, THREADS, 0, stream>>>(p, prod, agg, prior, out, N,
                                                    (s == 4) ? 1 : 0);
  }
}